// DCTResolution2D_60962765799520
// MI455X (gfx1250) — compile-verified
//
#include <hip/hip_runtime.h>
#include <math.h>

// ---------------------------------------------------------------------------
// out[b,c] = Mh (133x128) @ x[b,c] (128x128) @ Mw^T (128x133)
// where Mh/Mw fold DCT-II -> mask -> zero-pad -> DCT-III into one matrix.
// fp32 WMMA (v_wmma_f32_16x16x4_f32) on gfx1250, wave32.
// ---------------------------------------------------------------------------

typedef __attribute__((ext_vector_type(2))) float v2f;
typedef __attribute__((ext_vector_type(4))) float v4f;
typedef __attribute__((ext_vector_type(8))) float v8f;

#define HDIM    128
#define NEWDIM  133
#define NPAD    144          // 9 * 16, padded output rows/cols
#define MSTR    132          // LDS row stride (bank-conflict padding)
#define SMOOTHF 4.0f

// ---- kernel 1: build resampling matrices Mh, Mw : [NPAD x 128] each -------
__global__ void build_maps_kernel(const float* __restrict__ rate_weights,
                                  float* __restrict__ maps) {
    const int mat  = blockIdx.y;                       // 0 = Mh, 1 = Mw
    const int idx  = blockIdx.x * blockDim.x + threadIdx.x;  // 0..18431
    const int iout = idx >> 7;                         // 0..143
    const int iin  = idx & 127;                        // 0..127
    if (iout >= NPAD) return;

    float val = 0.0f;
    if (iout < NEWDIM) {
        // rate clamping:  min_allowed = max((1-4)/128, 0) = 0, max = 2
        float r = rate_weights[mat];
        r = fminf(fmaxf(r, 0.0f), 2.0f);
        const float crop = 128.0f * r;                 // new_smooth - SMOOTH

        const float PI     = 3.14159265358979323846f;
        const float s128   = sqrtf(2.0f / 128.0f);
        const float s133   = sqrtf(2.0f / 133.0f);
        const float isqrt2 = 0.70710678118654752440f;

        const float a_in  = PI * (2.0f * (float)iin  + 1.0f) / (2.0f * 128.0f);
        const float a_out = PI * (2.0f * (float)iout + 1.0f) / (2.0f * 133.0f);

        float acc = 0.0f;
        for (int u = 0; u < HDIM; ++u) {
            float mask = fminf(fmaxf((SMOOTHF + crop - (float)u) / SMOOTHF,
                                     0.0f), 1.0f);
            float scale0 = (u == 0) ? (isqrt2 * isqrt2) : 1.0f;  // both row-0 factors
            float d1 = cosf(a_in  * (float)u);   // DCT-II  (analysis,  N=128)
            float d2 = cosf(a_out * (float)u);   // DCT-III (synthesis, N=133)
            acc += scale0 * mask * d1 * d2;
        }
        val = acc * s128 * s133;
    }
    maps[(size_t)mat * (NPAD * HDIM) + iout * HDIM + iin] = val;
}

// ---- kernel 2: fused  out = Mh_panel @ X @ Mw^T  --------------------------
__global__ __launch_bounds__(256)
void dct_resample_kernel(const float* __restrict__ x,
                         const float* __restrict__ maps,
                         float* __restrict__ out) {
    __shared__ float MhL[48 * MSTR];   // 48-row panel of Mh
    __shared__ float Ts [48 * MSTR];   // T = Mh_panel @ X   (48 x 128)

    const int mu_g = blockIdx.x;               // row group: 0..2 (48 rows each)
    const int bc   = blockIdx.y;               // image: 0..2047
    const int tid  = threadIdx.x;
    const int wave = tid >> 5;                 // 0..7
    const int lane = tid & 31;
    const int nl   = lane & 15;                // N / M index inside fragment
    const int hi   = lane >> 4;                // half-wave select
    const int koff = hi << 1;                  // K offset {0,2} for A/B frags

    const float* __restrict__ Mh = maps;
    const float* __restrict__ Mw = maps + NPAD * HDIM;
    const float* __restrict__ X  = x + (size_t)bc * (HDIM * HDIM);

    // ---- stage Mh panel rows [mu_g*48, mu_g*48+48) into LDS ----
    {
        const float* src = Mh + (size_t)mu_g * 48 * HDIM;
        for (int i = tid; i < 48 * 32; i += 256) {
            const int r  = i >> 5;
            const int c4 = (i & 31) << 2;
            v4f v = *(const v4f*)(src + r * HDIM + c4);
            *(v4f*)(&MhL[r * MSTR + c4]) = v;
        }
    }
    __syncthreads();

    // ---- stage 1: Ts(48x128) = MhL(48x128) @ X(128x128) ----
    // wave w owns X column slab n0 = 16*w; 3 stacked 16x16 tiles (mu=0,1,2)
    {
        const int n0 = wave << 4;
        v8f acc0 = {}, acc1 = {}, acc2 = {};
        const float* xb = X + n0 + nl;
        for (int k0 = 0; k0 < HDIM; k0 += 4) {
            const float* xk = xb + (size_t)(k0 + koff) * HDIM;
            v2f b; b[0] = xk[0]; b[1] = xk[HDIM];          // B[k][n], B[k+1][n]
            const float* al = &MhL[nl * MSTR + k0 + koff];
            v2f a0 = *(const v2f*)(al);
            v2f a1 = *(const v2f*)(al + 16 * MSTR);
            v2f a2 = *(const v2f*)(al + 32 * MSTR);
            acc0 = __builtin_amdgcn_wmma_f32_16x16x4_f32(false, a0, false, b,
                                                         (short)0, acc0, false, false);
            acc1 = __builtin_amdgcn_wmma_f32_16x16x4_f32(false, a1, false, b,
                                                         (short)0, acc1, false, false);
            acc2 = __builtin_amdgcn_wmma_f32_16x16x4_f32(false, a2, false, b,
                                                         (short)0, acc2, false, false);
        }
        const int col = n0 + nl;
        const int rw0 = (hi << 3);                         // +8 for upper half
#pragma unroll
        for (int r = 0; r < 8; ++r) {
            Ts[( 0 + r + rw0) * MSTR + col] = acc0[r];
            Ts[(16 + r + rw0) * MSTR + col] = acc1[r];
            Ts[(32 + r + rw0) * MSTR + col] = acc2[r];
        }
    }
    __syncthreads();

    // ---- stage 2: out rows = Ts(48x128) @ Mw^T(128x144->133) ----
    const int rowbase = mu_g * 48;
    float* __restrict__ ob = out + (size_t)bc * (NEWDIM * NEWDIM);
    for (int jt = wave; jt < 9; jt += 8) {                 // wave-uniform
        const int j0 = jt << 4;
        v8f acc0 = {}, acc1 = {}, acc2 = {};
        const float* wb = Mw + (size_t)(j0 + nl) * HDIM;   // B[k][j] = Mw[j][k]
        for (int k0 = 0; k0 < HDIM; k0 += 4) {
            v2f b = *(const v2f*)(wb + k0 + koff);
            const float* al = &Ts[nl * MSTR + k0 + koff];
            v2f a0 = *(const v2f*)(al);
            v2f a1 = *(const v2f*)(al + 16 * MSTR);
            v2f a2 = *(const v2f*)(al + 32 * MSTR);
            acc0 = __builtin_amdgcn_wmma_f32_16x16x4_f32(false, a0, false, b,
                                                         (short)0, acc0, false, false);
            acc1 = __builtin_amdgcn_wmma_f32_16x16x4_f32(false, a1, false, b,
                                                         (short)0, acc1, false, false);
            acc2 = __builtin_amdgcn_wmma_f32_16x16x4_f32(false, a2, false, b,
                                                         (short)0, acc2, false, false);
        }
        const int jg = j0 + nl;
        if (jg < NEWDIM) {
            const int rw0 = rowbase + (hi << 3);
#pragma unroll
            for (int r = 0; r < 8; ++r) {
                const int m0 = rw0 + r;
                if (m0      < NEWDIM) ob[(size_t)(m0     ) * NEWDIM + jg] = acc0[r];
                if (m0 + 16 < NEWDIM) ob[(size_t)(m0 + 16) * NEWDIM + jg] = acc1[r];
                if (m0 + 32 < NEWDIM) ob[(size_t)(m0 + 32) * NEWDIM + jg] = acc2[r];
            }
        }
    }
}

extern "C" void kernel_launch(void* const* d_in, const int* in_sizes, int n_in,
                              void* d_out, int out_size, void* d_ws, size_t ws_size,
                              hipStream_t stream) {
    const float* x_dev  = (const float*)d_in[0];     // (32,64,128,128) fp32
    const float* rw_dev = (const float*)d_in[1];     // (2,) fp32
    float*       maps   = (float*)d_ws;              // 2 * 144*128 fp32 = 147456 B
    float*       o_dev  = (float*)d_out;             // (32,64,133,133) fp32

    // kernel 1: 2 matrices x 144x128 entries, one thread each
    dim3 g1(72, 2);
    build_maps_kernel<<<g1, 256, 0, stream>>>(rw_dev, maps);

    // kernel 2: 3 row-groups x 2048 images, 8 waves per block
    dim3 g2(3, 32 * 64);
    dct_resample_kernel<<<g2, 256, 0, stream>>>(x_dev, maps, o_dev);
}